// _BandSplit_34007551049836
// MI455X (gfx1250) — compile-verified
//
#include <hip/hip_runtime.h>

typedef __attribute__((ext_vector_type(2))) float v2f;
typedef __attribute__((ext_vector_type(8))) float v8f;

#define T_DIM 2000
#define FR    512
#define NBAND 32
#define NOUT  128

struct BandPtrs {
  const float* gamma[NBAND];
  const float* beta [NBAND];
  const float* W    [NBAND];
  const float* bias [NBAND];
};

// One block: one band x 16 consecutive (b,t) rows. 256 threads = 8 wave32.
// Each wave produces one 16x16 tile of the 16x128 output block via
// V_WMMA_F32_16X16X4_F32 (fp32 tensor op: full precision, BW-bound problem).
__global__ __launch_bounds__(256)
void bandsplit_ln_gemm(const float* __restrict__ x, float* __restrict__ out,
                       BandPtrs p)
{
  const int band = blockIdx.y;
  const int r0   = blockIdx.x * 16;        // flat row = b*T + t; T%16==0 -> b uniform
  const int b    = r0 / T_DIM;
  const int t0   = r0 - b * T_DIM;

  int d, dlog, start;
  if (band < 16)      { d = 16; dlog = 4; start = band * 16;            }
  else if (band < 24) { d = 32; dlog = 5; start = 256 + (band-16) * 32; }
  else                { d = 64; dlog = 6; start = 512 + (band-24) * 64; }

  const int tid  = threadIdx.x;
  const int r    = tid & 15;               // row within tile
  const int slot = tid >> 4;               // 0..15 column slot

  __shared__ float aTile[16][65];          // normalized A, padded (65 = 1 mod 64)
  __shared__ float wTile[64][129];         // W transposed: wTile[k][n], padded
  __shared__ float redS[16][16];
  __shared__ float redQ[16][16];
  __shared__ float rowMu[16], rowInv[16];

  // ---- stage W transposed into LDS (coalesced global reads) ----
  const float* __restrict__ W = p.W[band];
  for (int idx = tid; idx < (NOUT << dlog); idx += 256) {
    int n = idx >> dlog;                   // output channel
    int k = idx & (d - 1);                 // input channel
    wTile[k][n] = W[idx];
  }

  // ---- load x slice raw into aTile + per-thread partial moments ----
  const int nc = d >> 4;                   // columns per thread: 1 / 2 / 4
  float s = 0.f, q = 0.f;
  for (int i = 0; i < nc; ++i) {
    int c  = slot + (i << 4);
    int gc = start + c;
    int f  = gc >> 1, j = gc & 1;          // xf[b][t][2f+j] == x[b][f][t][j]
    float v = x[((b * FR + f) * T_DIM + (t0 + r)) * 2 + j];
    aTile[r][c] = v;
    s += v; q += v * v;
  }
  redS[r][slot] = s;
  redQ[r][slot] = q;
  __syncthreads();

  if (tid < 16) {
    float ss = 0.f, qq = 0.f;
    for (int k2 = 0; k2 < 16; ++k2) { ss += redS[tid][k2]; qq += redQ[tid][k2]; }
    float mu  = ss / (float)d;
    float var = qq / (float)d - mu * mu;
    rowMu[tid]  = mu;
    rowInv[tid] = rsqrtf(var + 1e-5f);
  }
  __syncthreads();

  // ---- LayerNorm in place (each thread owns its elements) ----
  {
    const float mu  = rowMu[r];
    const float inv = rowInv[r];
    const float* __restrict__ g  = p.gamma[band];
    const float* __restrict__ be = p.beta[band];
    for (int i = 0; i < nc; ++i) {
      int c = slot + (i << 4);
      aTile[r][c] = (aTile[r][c] - mu) * inv * g[c] + be[c];
    }
  }
  __syncthreads();

  // ---- GEMM: wave wv computes output columns [wv*16, wv*16+16) ----
  const int lane = tid & 31;
  const int wv   = tid >> 5;
  const int ln   = lane & 15;
  const int hi   = lane >> 4;              // 0|1 -> A/B K-offset 0|2, C/D M-offset 0|8
  const int n    = (wv << 4) + ln;

  const float bias = p.bias[band][n];      // bias is per-N: same across all 8 C VGPRs
  v8f acc;
  #pragma unroll
  for (int j = 0; j < 8; ++j) acc[j] = bias;

  for (int k = 0; k < d; k += 4) {
    const int kk = k + (hi << 1);
    v2f a, bb;
    a.x  = aTile[ln][kk];                  // A[M=ln][K=kk]   (16x4 fp32 layout)
    a.y  = aTile[ln][kk + 1];              // A[M=ln][K=kk+1]
    bb.x = wTile[kk][n];                   // B[K=kk][N=n] = W[n][kk]
    bb.y = wTile[kk + 1][n];
    acc = __builtin_amdgcn_wmma_f32_16x16x4_f32(
        /*neg_a=*/false, a, /*neg_b=*/false, bb,
        /*c_mod=*/(short)0, acc, /*reuse_a=*/false, /*reuse_b=*/false);
  }

  // ---- store D: out[((b*T+t)*32 + band)*128 + n], t = t0 + m, m = j + hi*8 ----
  #pragma unroll
  for (int j = 0; j < 8; ++j) {
    int m = j + (hi << 3);
    out[((r0 + m) * NBAND + band) * NOUT + n] = acc[j];
  }
}

extern "C" void kernel_launch(void* const* d_in, const int* in_sizes, int n_in,
                              void* d_out, int out_size, void* d_ws, size_t ws_size,
                              hipStream_t stream) {
  (void)in_sizes; (void)out_size; (void)d_ws; (void)ws_size;
  const float* x = (const float*)d_in[0];

  int dArr[NBAND];
  for (int i = 0; i < NBAND; ++i) dArr[i] = (i < 16) ? 16 : (i < 24 ? 32 : 64);

  BandPtrs p;
  if (n_in >= 1 + 4 * NBAND) {
    // dict tuples flattened into separate device arrays:
    // [x, g0..g31, b0..b31, W0..W31, bias0..bias31]
    for (int i = 0; i < NBAND; ++i) {
      p.gamma[i] = (const float*)d_in[1 + i];
      p.beta [i] = (const float*)d_in[1 + NBAND + i];
      p.W    [i] = (const float*)d_in[1 + 2 * NBAND + i];
      p.bias [i] = (const float*)d_in[1 + 3 * NBAND + i];
    }
  } else {
    // tuples delivered as concatenated flat arrays: [x, gcat, bcat, Wcat, biascat]
    const float* gcat = (const float*)d_in[1];
    const float* bcat = (const float*)d_in[2];
    const float* Wcat = (const float*)d_in[3];
    const float* icat = (const float*)d_in[4];
    long go = 0, Wo = 0, io = 0;
    for (int i = 0; i < NBAND; ++i) {
      p.gamma[i] = gcat + go;
      p.beta [i] = bcat + go;
      p.W    [i] = Wcat + Wo;
      p.bias [i] = icat + io;
      go += dArr[i];
      Wo += (long)NOUT * dArr[i];
      io += NOUT;
    }
  }

  dim3 grid((8 * T_DIM) / 16, NBAND);      // 1000 row tiles x 32 bands
  bandsplit_ln_gemm<<<grid, 256, 0, stream>>>(x, (float*)d_out, p);
}